// LightGCN_66907000537226
// MI455X (gfx1250) — compile-verified
//
#include <hip/hip_runtime.h>
#include <hip/hip_bf16.h>
#include <stdint.h>

#define NUM_USERS 200000
#define NUM_ITEMS 100000
#define N_NODES   (NUM_USERS + NUM_ITEMS)
#define EMB       64
#define TOTAL_ELEMS (N_NODES * EMB)

#define WAVES_PER_BLOCK 8
#define PIPE_DEPTH 8
#define SCAN_T 1024

// ---- CDNA5 async-to-LDS path detection (device pass only) ----
#if defined(__AMDGCN__) && __has_builtin(__builtin_amdgcn_global_load_async_to_lds_b64) && __has_builtin(__builtin_amdgcn_s_wait_asynccnt)
#define HAVE_ASYNC_LDS 1
#else
#define HAVE_ASYNC_LDS 0
#endif

// 64-bit payload type the async builtin expects: v2i32
typedef int lg_v2i __attribute__((vector_size(2 * sizeof(int))));
typedef __attribute__((address_space(1))) lg_v2i as1_v2i_t;
typedef __attribute__((address_space(3))) lg_v2i as3_v2i_t;

__device__ __forceinline__ as1_v2i_t* to_global_v2(const void* p) {
  return (as1_v2i_t*)(uintptr_t)p;
}
__device__ __forceinline__ as3_v2i_t* to_lds_v2(const void* p) {
  // generic shared addr: low 32 bits == LDS byte offset
  return (as3_v2i_t*)(uint32_t)(uintptr_t)p;
}

// ---------------- CSR build ----------------

__global__ void lg_zero(int* __restrict__ p, int n) {
  int i = blockIdx.x * blockDim.x + threadIdx.x;
  if (i < n) p[i] = 0;
}

__global__ void lg_count(const int* __restrict__ erow, int* __restrict__ cnt, int nnz) {
  int e = blockIdx.x * blockDim.x + threadIdx.x;
  if (e < nnz) atomicAdd(&cnt[erow[e]], 1);
}

// single-block exclusive scan over N_NODES counts -> row_ptr[0..N_NODES]
__global__ __launch_bounds__(SCAN_T) void lg_scan(const int* __restrict__ cnt,
                                                  int* __restrict__ row_ptr) {
  __shared__ int sm[SCAN_T];
  __shared__ int s_run;
  const int tid = threadIdx.x;
  if (tid == 0) s_run = 0;
  __syncthreads();
  for (int base = 0; base < N_NODES; base += SCAN_T) {
    const int i = base + tid;
    int v = (i < N_NODES) ? cnt[i] : 0;
    sm[tid] = v;
    __syncthreads();
    for (int off = 1; off < SCAN_T; off <<= 1) {
      int t = (tid >= off) ? sm[tid - off] : 0;
      __syncthreads();
      sm[tid] += t;
      __syncthreads();
    }
    const int incl  = sm[tid];
    const int total = sm[SCAN_T - 1];
    if (i < N_NODES) row_ptr[i] = s_run + incl - v;   // exclusive
    __syncthreads();
    if (tid == 0) s_run += total;
    __syncthreads();
  }
  if (tid == 0) row_ptr[N_NODES] = s_run;
}

__global__ void lg_fill(const int* __restrict__ erow, const int* __restrict__ ecol,
                        const float* __restrict__ eval, const int* __restrict__ row_ptr,
                        int* __restrict__ fill, int* __restrict__ csr_col,
                        float* __restrict__ csr_val, int nnz) {
  int e = blockIdx.x * blockDim.x + threadIdx.x;
  if (e < nnz) {
    const int r = erow[e];
    const int pos = row_ptr[r] + atomicAdd(&fill[r], 1);
    csr_col[pos] = ecol[e];
    csr_val[pos] = eval[e];
  }
}

// ---------------- embedding init / finalize ----------------

__global__ void lg_init(const float* __restrict__ user_emb, const float* __restrict__ item_emb,
                        float* __restrict__ h0, float* __restrict__ acc) {
  int i = blockIdx.x * blockDim.x + threadIdx.x;
  if (i < TOTAL_ELEMS) {
    const float v = (i < NUM_USERS * EMB) ? user_emb[i] : item_emb[i - NUM_USERS * EMB];
    h0[i] = v;
    acc[i] = v;
  }
}

__global__ void lg_scale(float* __restrict__ acc) {
  int i = blockIdx.x * blockDim.x + threadIdx.x;
  if (i < TOTAL_ELEMS) acc[i] *= 0.25f;  // 1/(N_LAYERS+1)
}

// ---------------- pull-based SpMM: one wave32 per row ----------------
// Each lane owns columns [2*lane, 2*lane+1]. Neighbor rows (256B) are
// gathered via CDNA5 async-to-LDS DMA, pipelined PIPE_DEPTH deep on ASYNCcnt.
// Row index is forced uniform (readfirstlane) so CSR metadata moves on the
// scalar pipe (s_load) and the vector memory path carries only the gathers.

__global__ __launch_bounds__(32 * WAVES_PER_BLOCK) void lg_spmm(
    const int* __restrict__ row_ptr, const int* __restrict__ csr_col,
    const float* __restrict__ csr_val, const float* __restrict__ x,
    float* __restrict__ h_out, float* __restrict__ acc) {
  const int lane = threadIdx.x & 31;
  const int row  = __builtin_amdgcn_readfirstlane(
      blockIdx.x * WAVES_PER_BLOCK + (threadIdx.x >> 5));
  if (row >= N_NODES) return;
  const int w = __builtin_amdgcn_readfirstlane(threadIdx.x >> 5);

  const int e0 = row_ptr[row];
  const int e1 = row_ptr[row + 1];
  const int nE = e1 - e0;
  float a0 = 0.f, a1 = 0.f;

#if HAVE_ASYNC_LDS
  __shared__ __align__(16) float stage[WAVES_PER_BLOCK][PIPE_DEPTH][EMB];
  // prologue: fill pipeline
  const int pro = nE < PIPE_DEPTH ? nE : PIPE_DEPTH;
  for (int k = 0; k < pro; ++k) {
    const int c = csr_col[e0 + k];
    const float* gsrc = x + (size_t)c * EMB + lane * 2;  // 8B/lane, 256B/wave
    __builtin_amdgcn_global_load_async_to_lds_b64(
        to_global_v2(gsrc), to_lds_v2(&stage[w][k][lane * 2]), 0, 0);
  }
  const int main_end = nE > PIPE_DEPTH ? nE - PIPE_DEPTH : 0;
  int j = 0;
  for (; j < main_end; ++j) {
    // pipeline full: exactly PIPE_DEPTH outstanding -> oldest (j) is done
    __builtin_amdgcn_s_wait_asynccnt(PIPE_DEPTH - 1);
    const float v   = csr_val[e0 + j];
    const int slot  = j & (PIPE_DEPTH - 1);
    const float2 ss = *(const float2*)&stage[w][slot][lane * 2];  // ds_load_b64
    a0 = fmaf(v, ss.x, a0);
    a1 = fmaf(v, ss.y, a1);
    const int c = csr_col[e0 + j + PIPE_DEPTH];
    const float* gsrc = x + (size_t)c * EMB + lane * 2;
    __builtin_amdgcn_global_load_async_to_lds_b64(
        to_global_v2(gsrc), to_lds_v2(&stage[w][slot][lane * 2]), 0, 0);
  }
  __builtin_amdgcn_s_wait_asynccnt(0);  // drain
  for (; j < nE; ++j) {
    const float v   = csr_val[e0 + j];
    const int slot  = j & (PIPE_DEPTH - 1);
    const float2 ss = *(const float2*)&stage[w][slot][lane * 2];
    a0 = fmaf(v, ss.x, a0);
    a1 = fmaf(v, ss.y, a1);
  }
#else
  for (int j = 0; j < nE; ++j) {
    const int c   = csr_col[e0 + j];
    const float v = csr_val[e0 + j];
    const float2 xx = *(const float2*)(x + (size_t)c * EMB + lane * 2);
    a0 = fmaf(v, xx.x, a0);
    a1 = fmaf(v, xx.y, a1);
  }
#endif

  const int o = row * EMB + lane * 2;
  h_out[o]     = a0;
  h_out[o + 1] = a1;
  acc[o]     += a0;   // each (row,lane) pair is unique: no races
  acc[o + 1] += a1;
}

// ---------------- host orchestration ----------------

extern "C" void kernel_launch(void* const* d_in, const int* in_sizes, int n_in,
                              void* d_out, int out_size, void* d_ws, size_t ws_size,
                              hipStream_t stream) {
  const float* user_emb = (const float*)d_in[0];
  const float* item_emb = (const float*)d_in[1];
  const int*   erow     = (const int*)d_in[2];
  const int*   ecol     = (const int*)d_in[3];
  const float* eval     = (const float*)d_in[4];
  const int    nnz      = in_sizes[2];

  float* acc = (float*)d_out;  // [N_NODES * EMB]

  char* ws = (char*)d_ws;
  size_t off = 0;
  auto alloc = [&](size_t bytes) -> void* {
    off = (off + 255) & ~(size_t)255;
    void* p = ws + off;
    off += bytes;
    return p;
  };
  int*   cnt     = (int*)  alloc(sizeof(int)   * N_NODES);
  int*   fill    = (int*)  alloc(sizeof(int)   * N_NODES);
  int*   row_ptr = (int*)  alloc(sizeof(int)   * (N_NODES + 1));
  int*   csr_col = (int*)  alloc(sizeof(int)   * (size_t)nnz);
  float* csr_val = (float*)alloc(sizeof(float) * (size_t)nnz);
  float* h0      = (float*)alloc(sizeof(float) * (size_t)TOTAL_ELEMS);
  float* h1      = (float*)alloc(sizeof(float) * (size_t)TOTAL_ELEMS);
  (void)ws_size; (void)n_in; (void)out_size;

  const int tb = 256;
  // CSR build (once; reused for all 3 layers)
  lg_zero <<<(N_NODES + tb - 1) / tb, tb, 0, stream>>>(cnt, N_NODES);
  lg_zero <<<(N_NODES + tb - 1) / tb, tb, 0, stream>>>(fill, N_NODES);
  lg_count<<<(nnz + tb - 1) / tb, tb, 0, stream>>>(erow, cnt, nnz);
  lg_scan <<<1, SCAN_T, 0, stream>>>(cnt, row_ptr);
  lg_fill <<<(nnz + tb - 1) / tb, tb, 0, stream>>>(erow, ecol, eval, row_ptr, fill,
                                                   csr_col, csr_val, nnz);
  // ego embeddings -> h0 and acc
  lg_init <<<(TOTAL_ELEMS + tb - 1) / tb, tb, 0, stream>>>(user_emb, item_emb, h0, acc);

  // 3 propagation layers, accumulating into acc
  const int spmm_grid = (N_NODES + WAVES_PER_BLOCK - 1) / WAVES_PER_BLOCK;
  const int spmm_tb   = 32 * WAVES_PER_BLOCK;
  lg_spmm<<<spmm_grid, spmm_tb, 0, stream>>>(row_ptr, csr_col, csr_val, h0, h1, acc);
  lg_spmm<<<spmm_grid, spmm_tb, 0, stream>>>(row_ptr, csr_col, csr_val, h1, h0, acc);
  lg_spmm<<<spmm_grid, spmm_tb, 0, stream>>>(row_ptr, csr_col, csr_val, h0, h1, acc);

  lg_scale<<<(TOTAL_ELEMS + tb - 1) / tb, tb, 0, stream>>>(acc);
}